// RedNet_FC_34024730919078
// MI455X (gfx1250) — compile-verified
//
#include <hip/hip_runtime.h>
#include <hip/hip_bf16.h>

typedef __attribute__((ext_vector_type(2))) float v2f;
typedef __attribute__((ext_vector_type(8))) float v8f;

#define HWN 3136
#define HDIM 56
#define WDIM 56
#define BATCH 8

// ---------------------------------------------------------------------------
// GEMM via V_WMMA_F32_16X16X4_F32:  out[b,o,n] = sum_k w[o,k] * act(in[b,k,n])
// act = optional fused BN(scale,shift)+ReLU on the B-fragment loads (BN param).
// One wave computes one 16-row x 64-pixel strip = 4 independent 16x16 tiles
// (4 acc chains hide the WMMA accumulation latency; A fragment reused 4x).
// MASK variant handles Cout not a multiple of 16 (span conv, O=196) by
// clamping the A row index and zeroing the value (no divergent-EXEC loads).
// ---------------------------------------------------------------------------
template <bool BN, bool MASK>
__global__ __launch_bounds__(256) void gemm1x1_wmma(
    const float* __restrict__ in, const float* __restrict__ w,
    float* __restrict__ out,
    const float* __restrict__ scale, const float* __restrict__ shift,
    int Cin, int Cout, int oTiles)
{
    const int pGroups = HWN / 64;                // 49 groups of 4 pixel-tiles
    const int perBatch = oTiles * pGroups;
    const int wave = threadIdx.x >> 5;
    const int lane = threadIdx.x & 31;
    const int tile = blockIdx.x * 8 + wave;
    if (tile >= BATCH * perBatch) return;        // wave-uniform

    const int b  = tile / perBatch;
    const int t  = tile % perBatch;
    const int ot = t / pGroups;
    const int pg = t % pGroups;

    const int hi = lane >> 4;                    // half-wave: K offset +2
    const int l  = lane & 15;
    const int m  = ot * 16 + l;                  // A row (output channel)
    const int mc = MASK ? min(m, Cout - 1) : m;  // clamped (always in-bounds)
    const float aval = (!MASK || m < Cout) ? 1.0f : 0.0f;

    const float* __restrict__ ap  = w + (size_t)mc * Cin + (hi << 1);
    const float* __restrict__ bp  = in + (size_t)b * Cin * HWN
                                       + (size_t)(hi << 1) * HWN + pg * 64 + l;
    const float* __restrict__ scp = scale + (hi << 1);
    const float* __restrict__ shp = shift + (hi << 1);

    v8f acc[4] = {{}, {}, {}, {}};

    for (int k0 = 0; k0 < Cin; k0 += 4) {
        v2f a = *(const v2f*)ap;  ap += 4;       // global_load_b64
        if (MASK) { a.x *= aval; a.y *= aval; }
        v2f sc2, sh2;
        if constexpr (BN) {
            sc2 = *(const v2f*)scp;  scp += 4;
            sh2 = *(const v2f*)shp;  shp += 4;
        }
        #pragma unroll
        for (int tt = 0; tt < 4; ++tt) {
            float b0 = bp[tt * 16];
            float b1 = bp[tt * 16 + HWN];
            if constexpr (BN) {
                b0 = fmaxf(fmaf(b0, sc2.x, sh2.x), 0.0f);
                b1 = fmaxf(fmaf(b1, sc2.y, sh2.y), 0.0f);
            }
            v2f bf; bf.x = b0; bf.y = b1;
            acc[tt] = __builtin_amdgcn_wmma_f32_16x16x4_f32(
                          false, a, false, bf, (short)0, acc[tt], false, false);
        }
        bp += (size_t)4 * HWN;
    }

    float* __restrict__ outb = out + (size_t)b * Cout * HWN + pg * 64 + l;
    const int mbase = ot * 16 + (hi ? 8 : 0);
    #pragma unroll
    for (int tt = 0; tt < 4; ++tt) {
        #pragma unroll
        for (int v = 0; v < 8; ++v) {
            const int mo = mbase + v;
            if (!MASK || mo < Cout)
                outb[(size_t)mo * HWN + tt * 16] = acc[tt][v];
        }
    }
}

// ---------------------------------------------------------------------------
// Training-mode BN statistics: one block per channel, reduce over B*H*W.
// Emits fused affine: scale[c] = g*rsqrt(var+eps), shift[c] = b - mean*scale.
// ---------------------------------------------------------------------------
__global__ __launch_bounds__(256) void bn_stats(
    const float* __restrict__ t, const float* __restrict__ gamma,
    const float* __restrict__ beta,
    float* __restrict__ scale, float* __restrict__ shift, int C)
{
    __shared__ float ssum[256];
    __shared__ float ssq[256];
    const int c = blockIdx.x;
    float s = 0.0f, q = 0.0f;
    for (int b = 0; b < BATCH; ++b) {
        const float* __restrict__ p = t + ((size_t)b * C + c) * HWN;
        for (int i = threadIdx.x; i < HWN; i += 256) {
            float v = p[i];
            s += v; q += v * v;
        }
    }
    ssum[threadIdx.x] = s; ssq[threadIdx.x] = q;
    __syncthreads();
    for (int off = 128; off > 0; off >>= 1) {
        if (threadIdx.x < off) {
            ssum[threadIdx.x] += ssum[threadIdx.x + off];
            ssq[threadIdx.x]  += ssq[threadIdx.x + off];
        }
        __syncthreads();
    }
    if (threadIdx.x == 0) {
        const float N = (float)(BATCH * HWN);
        const float mean = ssum[0] / N;
        const float var  = ssq[0] / N - mean * mean;
        const float sc   = gamma[c] * rsqrtf(var + 1e-5f);
        scale[c] = sc;
        shift[c] = beta[c] - mean * sc;
    }
}

// ---------------------------------------------------------------------------
// Involution stencil: t3[b,c,y,x] = sum_{k<49} wk[b, (c/16)*49+k, y, x] *
//                                   relu(bn1(t1[b,c,y+dy,x+dx]))   (zero pad)
// ---------------------------------------------------------------------------
__global__ __launch_bounds__(256) void involution7(
    const float* __restrict__ t1, const float* __restrict__ wk,
    const float* __restrict__ scale1, const float* __restrict__ shift1,
    float* __restrict__ t3, int C)
{
    const int idx = blockIdx.x * 256 + threadIdx.x;
    const int total = BATCH * C * HWN;
    if (idx >= total) return;
    const int x = idx % WDIM;
    const int y = (idx / WDIM) % HDIM;
    const int c = (idx / HWN) % C;
    const int b = idx / (HWN * C);
    const int g = c >> 4;                         // GC = 16
    const float sc = scale1[c], sh = shift1[c];

    const float* __restrict__ tin = t1 + ((size_t)b * C + c) * HWN;
    const float* __restrict__ wkp = wk + ((size_t)b * 196 + (size_t)g * 49) * HWN
                                       + (size_t)y * WDIM + x;
    float acc = 0.0f;
    int k = 0;
    #pragma unroll
    for (int dy = -3; dy <= 3; ++dy) {
        const int yy = y + dy;
        #pragma unroll
        for (int dx = -3; dx <= 3; ++dx, ++k) {
            const int xx = x + dx;
            if (yy >= 0 && yy < HDIM && xx >= 0 && xx < WDIM) {
                const float v = fmaxf(fmaf(tin[yy * WDIM + xx], sc, sh), 0.0f);
                acc = fmaf(wkp[(size_t)k * HWN], v, acc);
            }
        }
    }
    t3[idx] = acc;
}

// ---------------------------------------------------------------------------
// Final fusion: out = relu( bn3(t4) + x )
// ---------------------------------------------------------------------------
__global__ __launch_bounds__(256) void final_fuse(
    const float* __restrict__ t4, const float* __restrict__ xres,
    const float* __restrict__ scale4, const float* __restrict__ shift4,
    float* __restrict__ out, int C)
{
    const int idx = blockIdx.x * 256 + threadIdx.x;
    const int total = BATCH * C * HWN;
    if (idx >= total) return;
    const int c = (idx / HWN) % C;
    out[idx] = fmaxf(fmaf(t4[idx], scale4[c], shift4[c]) + xres[idx], 0.0f);
}

extern "C" void kernel_launch(void* const* d_in, const int* in_sizes, int n_in,
                              void* d_out, int out_size, void* d_ws, size_t ws_size,
                              hipStream_t stream) {
    (void)in_sizes; (void)n_in; (void)out_size; (void)ws_size;
    const float* x  = (const float*)d_in[0];
    const float* w1 = (const float*)d_in[1];
    const float* g1 = (const float*)d_in[2];
    const float* b1 = (const float*)d_in[3];
    const float* wr = (const float*)d_in[4];
    const float* gr = (const float*)d_in[5];
    const float* br = (const float*)d_in[6];
    const float* wsp= (const float*)d_in[7];
    const float* g2 = (const float*)d_in[8];
    const float* b2 = (const float*)d_in[9];
    const float* w3 = (const float*)d_in[10];
    const float* g3 = (const float*)d_in[11];
    const float* b3 = (const float*)d_in[12];
    float* out = (float*)d_out;

    // Workspace layout (floats). t4 reuses the t1/t2/wk region (dead by then).
    float* wsf = (float*)d_ws;
    float* t1  = wsf;                       // 8*64*3136  = 1,605,632
    float* t2  = wsf + 1605632;             // 8*16*3136  =   401,408
    float* wkb = wsf + 2007040;             // 8*196*3136 = 4,917,248
    float* t3  = wsf + 6924288;             // 8*64*3136  = 1,605,632
    float* t4  = wsf;                       // 8*256*3136 = 6,422,528 (reuse)
    float* st  = wsf + 8529920;             // stats (8-byte aligned pairs)
    float *sc1 = st,       *sh1 = st + 64;
    float *sc2 = st + 128, *sh2 = st + 144;
    float *sc3 = st + 160, *sh3 = st + 224;
    float *sc4 = st + 288, *sh4 = st + 544;

    const int pGroups = HWN / 64;           // 49

    // 1) conv1: 256 -> 64 (raw)
    {
        int oT = 4, tiles = BATCH * oT * pGroups;    // 1568 waves
        gemm1x1_wmma<false, false><<<tiles / 8, 256, 0, stream>>>(
            x, w1, t1, nullptr, nullptr, 256, 64, oT);
    }
    // 2) BN1 stats
    bn_stats<<<64, 256, 0, stream>>>(t1, g1, b1, sc1, sh1, 64);

    // 3) reduce conv: relu(bn1(t1)) -> 16 channels (raw)
    {
        int oT = 1, tiles = BATCH * oT * pGroups;    // 392 waves
        gemm1x1_wmma<true, false><<<tiles / 8, 256, 0, stream>>>(
            t1, wr, t2, sc1, sh1, 64, 16, oT);
    }
    // 4) BN-reduce stats
    bn_stats<<<16, 256, 0, stream>>>(t2, gr, br, sc2, sh2, 16);

    // 5) span conv: relu(bn(t2)) -> 196 per-pixel kernel channels (raw)
    {
        int oT = 13, tiles = BATCH * oT * pGroups;   // 5096 waves (masked rows)
        gemm1x1_wmma<true, true><<<tiles / 8, 256, 0, stream>>>(
            t2, wsp, wkb, sc2, sh2, 16, 196, oT);
    }

    // 6) involution stencil (fuses bn1+relu on its input loads)
    {
        int total = BATCH * 64 * HWN;                // 1,605,632
        involution7<<<(total + 255) / 256, 256, 0, stream>>>(
            t1, wkb, sc1, sh1, t3, 64);
    }
    // 7) BN2 stats
    bn_stats<<<64, 256, 0, stream>>>(t3, g2, b2, sc3, sh3, 64);

    // 8) conv3: relu(bn2(t3)) -> 256 channels (raw)   [t4 overlaps dead t1/t2/wk]
    {
        int oT = 16, tiles = BATCH * oT * pGroups;   // 6272 waves
        gemm1x1_wmma<true, false><<<tiles / 8, 256, 0, stream>>>(
            t3, w3, t4, sc3, sh3, 64, 256, oT);
    }
    // 9) BN3 stats
    bn_stats<<<256, 256, 0, stream>>>(t4, g3, b3, sc4, sh4, 256);

    // 10) out = relu(bn3(t4) + x)
    {
        int total = BATCH * 256 * HWN;               // 6,422,528
        final_fuse<<<(total + 255) / 256, 256, 0, stream>>>(
            t4, x, sc4, sh4, out, 256);
    }
}